// gpu_iter_cal_grad_47090021433586
// MI455X (gfx1250) — compile-verified
//
#include <hip/hip_runtime.h>
#include <math.h>

typedef __attribute__((ext_vector_type(2))) float v2f;
typedef __attribute__((ext_vector_type(8))) float v8f;

// ---- workspace layout (floats) ----
#define WS_INV_A   0
#define WS_INV_B   64
#define WS_TWOB2   128
#define WS_RHO     192
#define WS_IRHO    256
#define WS_RHO2    320
#define WS_IRHO2   384
#define WS_D       448
#define WS_OMEGA   512           // 512 entries
#define WS_RNG     1024          // 512 entries
#define WS_MNORM   1536          // 1 entry

__device__ __forceinline__ float frcp(float x)  { return __builtin_amdgcn_rcpf(x); }
__device__ __forceinline__ float fsqrt(float x) { return __builtin_amdgcn_sqrtf(x); }

// CPS 'var' branch: P/SV eigenfunction pieces with evanescent scaling.
__device__ __forceinline__ void branch_fn(float pp, float rr, float wvno, float xk,
                                          float dpth, float& c, float& w, float& x,
                                          float& pex)
{
    bool lt = wvno < xk;
    bool eq = wvno == xk;
    float rr_safe = (rr == 0.0f) ? 1.0f : rr;
    float s   = __sinf(pp);
    float fac = (pp < 16.0f) ? __expf(-2.0f * pp) : 0.0f;
    float c_ev = (1.0f + fac) * 0.5f;
    float s_ev = (1.0f - fac) * 0.5f;
    pex = (lt || eq) ? 0.0f : pp;
    float inv_rr = frcp(rr_safe);
    c = lt ? __cosf(pp) : (eq ? 1.0f : c_ev);
    w = lt ? (s * inv_rr) : (eq ? dpth : s_ev * inv_rr);
    x = lt ? (-rr * s) : (eq ? 0.0f : rr * s_ev);
}

// Half-space E row-vector.
__device__ __forceinline__ void halfspace_e(float wvno, float wvno2, float xka,
                                            float xkb, float gammk, float r,
                                            float& e0, float& e1, float& e2,
                                            float& e3, float& e4)
{
    float ra = fsqrt((wvno + xka) * fabsf(wvno - xka));
    float rb = fsqrt((wvno + xkb) * fabsf(wvno - xkb));
    float gam = gammk * wvno2;
    float gamm1 = gam - 1.0f;
    e0 = r * r * (gamm1 * gamm1 - gam * gammk * ra * rb);
    e1 = -r * ra;
    e2 = r * (gamm1 - gammk * ra * rb);
    e3 = r * rb;
    e4 = wvno2 - ra * rb;
}

// One Dunkin layer step: e^T <- normalize(e^T * CA(layer)).
__device__ __forceinline__ void layer_step(float wvno, float wvno2,
                                           float xka, float xkb, float gammk,
                                           float dm, float rm, float irho,
                                           float rho2, float irho2,
                                           float& e0, float& e1, float& e2,
                                           float& e3, float& e4)
{
    float gam = gammk * wvno2;
    float ra = fsqrt((wvno + xka) * fabsf(wvno - xka));
    float rb = fsqrt((wvno + xkb) * fabsf(wvno - xkb));
    float p = ra * dm;
    float q = rb * dm;

    float cosp, w, x, pex; branch_fn(p, ra, wvno, xka, dm, cosp, w, x, pex);
    float cosq, y, z, sex; branch_fn(q, rb, wvno, xkb, dm, cosq, y, z, sex);
    float exa = pex + sex;
    float a0 = (exa < 60.0f) ? __expf(-exa) : 0.0f;
    float cpcq = cosp * cosq, cpy = cosp * y, cpz = cosp * z;
    float cqw = cosq * w, cqx = cosq * x;
    float xy = x * y, xz = x * z, wy = w * y, wz = w * z;

    float gamm1 = gam - 1.0f;
    float twgm1 = gam + gamm1;
    float gmgmk = gam * gammk;
    float gmgm1 = gam * gamm1;
    float gm1sq = gamm1 * gamm1;
    float a0pq = a0 - cpcq;

    float c00 = cpcq - 2.0f * gmgm1 * a0pq - gmgmk * xz - wvno2 * gm1sq * wy;
    float c01 = (wvno2 * cpy - cqx) * irho;
    float c02 = -(twgm1 * a0pq + gammk * xz + wvno2 * gamm1 * wy) * irho;
    float c03 = (cpz - wvno2 * cqw) * irho;
    float c04 = -(2.0f * wvno2 * a0pq + xz + wvno2 * wvno2 * wy) * irho2;
    float c10 = (gmgmk * cpz - gm1sq * cqw) * rm;
    float c11 = cpcq;
    float c12 = gammk * cpz - gamm1 * cqw;
    float c13 = -wz;
    float c30 = (gm1sq * cpy - gmgmk * cqx) * rm;
    float c31 = -xy;
    float c32 = gamm1 * cpy - gammk * cqx;
    float c40 = -(2.0f * gmgmk * gm1sq * a0pq + gmgmk * gmgmk * xz
                  + gm1sq * gm1sq * wy) * rho2;
    float c42 = -(gmgm1 * twgm1 * a0pq + gam * gammk * gammk * xz
                  + gamm1 * gm1sq * wy) * rm;
    float tt = -2.0f * wvno2;
    float c20 = tt * c42;
    float c21 = tt * c32;
    float c22 = a0 + 2.0f * (cpcq - c00);
    float c23 = tt * c12;
    float c24 = tt * c02;

    float ee0 = e0 * c00 + e1 * c10 + e2 * c20 + e3 * c30 + e4 * c40;
    float ee1 = e0 * c01 + e1 * c11 + e2 * c21 + e3 * c31 + e4 * c30;
    float ee2 = e0 * c02 + e1 * c12 + e2 * c22 + e3 * c32 + e4 * c42;
    float ee3 = e0 * c03 + e1 * c13 + e2 * c23 + e3 * c11 + e4 * c10;
    float ee4 = e0 * c04 + e1 * c03 + e2 * c24 + e3 * c01 + e4 * c00;

    float t1 = fmaxf(fmaxf(fabsf(ee0), fabsf(ee1)),
                     fmaxf(fmaxf(fabsf(ee2), fabsf(ee3)), fabsf(ee4)));
    t1 = (t1 < 1.0e-30f) ? 1.0f : t1;
    float it = frcp(t1);
    e0 = ee0 * it; e1 = ee1 * it; e2 = ee2 * it; e3 = ee3 * it; e4 = ee4 * it;
}

// Specialized: omega uniform per block, xka/xkb/gammk pre-hoisted into LDS.
__device__ float dltar4_pre(float wvno, int nL,
                            const float* __restrict__ xka,
                            const float* __restrict__ xkb,
                            const float* __restrict__ gmk,
                            const float* __restrict__ rho,
                            const float* __restrict__ irho,
                            const float* __restrict__ rho2,
                            const float* __restrict__ irho2,
                            const float* __restrict__ dd)
{
    float wvno2 = wvno * wvno;
    int h = nL - 1;
    float e0, e1, e2, e3, e4;
    halfspace_e(wvno, wvno2, xka[h], xkb[h], gmk[h], rho[h], e0, e1, e2, e3, e4);
    for (int m = nL - 2; m >= 0; --m)
        layer_step(wvno, wvno2, xka[m], xkb[m], gmk[m], dd[m], rho[m], irho[m],
                   rho2[m], irho2[m], e0, e1, e2, e3, e4);
    return e0;
}

// Generic: omega per-thread.
__device__ float dltar4_gen(float wvno, float omega, int nL,
                            const float* __restrict__ ia,
                            const float* __restrict__ ib,
                            const float* __restrict__ t2b2,
                            const float* __restrict__ rho,
                            const float* __restrict__ irho,
                            const float* __restrict__ rho2,
                            const float* __restrict__ irho2,
                            const float* __restrict__ dd)
{
    omega = fmaxf(omega, 1.0e-4f);
    float inv_om2 = frcp(omega * omega);
    float wvno2 = wvno * wvno;
    int h = nL - 1;
    float e0, e1, e2, e3, e4;
    halfspace_e(wvno, wvno2, omega * ia[h], omega * ib[h], t2b2[h] * inv_om2,
                rho[h], e0, e1, e2, e3, e4);
    for (int m = nL - 2; m >= 0; --m)
        layer_step(wvno, wvno2, omega * ia[m], omega * ib[m], t2b2[m] * inv_om2,
                   dd[m], rho[m], irho[m], rho2[m], irho2[m], e0, e1, e2, e3, e4);
    return e0;
}

// Kernel 1: layer constants + omega
__global__ void prep_kernel(const float* __restrict__ tlist,
                            const float* __restrict__ b,
                            const float* __restrict__ d,
                            float* __restrict__ ws, int nL, int nT)
{
    for (int i = threadIdx.x; i < nT; i += blockDim.x)
        ws[WS_OMEGA + i] = 6.283185307179586f / tlist[i];
    for (int i = threadIdx.x; i < nL; i += blockDim.x) {
        float bb = b[i];
        float b2 = bb * bb;
        float a = 0.9409f + 2.0947f * bb - 0.8206f * b2 + 0.2683f * b2 * bb
                  - 0.0251f * b2 * b2;
        float a2 = a * a;
        float rho = 1.6612f * a - 0.4721f * a2 + 0.0671f * a2 * a
                    - 0.0043f * a2 * a2 + 0.000106f * a2 * a2 * a;
        float irho = 1.0f / rho;
        ws[WS_INV_A + i] = 1.0f / a;
        ws[WS_INV_B + i] = 1.0f / bb;
        ws[WS_TWOB2 + i] = 2.0f * b2;
        ws[WS_RHO + i]   = rho;
        ws[WS_IRHO + i]  = irho;
        ws[WS_RHO2 + i]  = rho * rho;
        ws[WS_IRHO2 + i] = irho * irho;
        ws[WS_D + i]     = d[i];
    }
}

// Kernel 2: m_norm = sum |DAMP*(L@b)/nL| via V_WMMA_F32_16X16X4_F32 (one wave).
// L staged zero-padded in LDS (clamped-address loads, no exec-branchy guards);
// b broadcast across all 16 B-columns -> every D column holds L@b.
__global__ void wmma_matvec_kernel(const float* __restrict__ L,
                                   const float* __restrict__ b,
                                   float* __restrict__ ws, int nL)
{
    __shared__ float Ls[64 * 64];
    __shared__ float bs[64];
    __shared__ float parts[2];
    int lane = threadIdx.x;            // 0..31, wave32
    int halfsel = lane >> 4;           // 0: lanes 0-15, 1: lanes 16-31
    int mrow = lane & 15;

    {   // b, zero-padded to 64 (clamped address, select)
        float b0 = b[(lane < nL) ? lane : 0];
        float b1 = b[(lane + 32 < nL) ? (lane + 32) : 0];
        bs[lane]      = (lane < nL)      ? b0 : 0.0f;
        bs[lane + 32] = (lane + 32 < nL) ? b1 : 0.0f;
    }
    // Stage L (nL x nL) into 64x64 zero-padded LDS tile; unconditional loads.
    for (int idx = lane; idx < 64 * 64; idx += 32) {
        int row = idx >> 6, col = idx & 63;
        bool ok = (row < nL) & (col < nL);
        float v = L[ok ? (row * nL + col) : 0];
        Ls[idx] = ok ? v : 0.0f;
    }
    __syncthreads();

    float msum = 0.0f;
    for (int mt = 0; mt < 4; ++mt) {
        int row = mt * 16 + mrow;      // A-matrix M index for this lane
        v8f c = {0.f, 0.f, 0.f, 0.f, 0.f, 0.f, 0.f, 0.f};
#pragma unroll
        for (int kc = 0; kc < 13; ++kc) {
            // A 16x4 f32 layout: lanes 0-15 hold K0/K1, lanes 16-31 hold K2/K3
            int ka = kc * 4 + halfsel * 2;             // even -> 8B aligned
            v2f av = *(const v2f*)&Ls[row * 64 + ka];  // ds_load_b64
            v2f bv = *(const v2f*)&bs[ka];             // K striping, N-broadcast
            c = __builtin_amdgcn_wmma_f32_16x16x4_f32(
                    false, av, false, bv, (short)0, c, false, false);
        }
        // D layout: lanes 0-15 VGPR r -> M=r; lanes 16-31 -> M=r+8. Column N=0.
        if ((lane & 15) == 0) {
#pragma unroll
            for (int r = 0; r < 8; ++r) {
                int orow = mt * 16 + halfsel * 8 + r;
                if (orow < nL) msum += fabsf(0.5f * c[r] / (float)nL);
            }
        }
    }
    if (lane == 0)  parts[0] = msum;
    if (lane == 16) parts[1] = msum;
    __syncthreads();
    if (lane == 0) ws[WS_MNORM] = parts[0] + parts[1];
}

// Kernel 3: per-period range of secular function over trial velocities.
__global__ void det_kernel(const float* __restrict__ Clist,
                           float* __restrict__ ws, int nL, int nC)
{
    __shared__ float s_xka[64], s_xkb[64], s_gmk[64];
    __shared__ float s_rho[64], s_irho[64], s_rho2[64], s_irho2[64], s_d[64];
    __shared__ float red_min[256], red_max[256];
    int tid = threadIdx.x;
    int t = blockIdx.x;
    __builtin_prefetch(&ws[WS_INV_A], 0, 0);   // global_prefetch_b8

    float omega = ws[WS_OMEGA + t];            // block-uniform
    float om_c = fmaxf(omega, 1.0e-4f);
    float inv_om2 = frcp(om_c * om_c);
    for (int i = tid; i < nL; i += blockDim.x) {
        s_xka[i]   = om_c * ws[WS_INV_A + i];
        s_xkb[i]   = om_c * ws[WS_INV_B + i];
        s_gmk[i]   = ws[WS_TWOB2 + i] * inv_om2;
        s_rho[i]   = ws[WS_RHO + i];
        s_irho[i]  = ws[WS_IRHO + i];
        s_rho2[i]  = ws[WS_RHO2 + i];
        s_irho2[i] = ws[WS_IRHO2 + i];
        s_d[i]     = ws[WS_D + i];
    }
    __syncthreads();

    float lmin = 3.402823466e38f, lmax = -3.402823466e38f;
    for (int c = tid; c < nC; c += blockDim.x) {
        __builtin_prefetch(&Clist[c + blockDim.x], 0, 0);
        float wv = omega * frcp(Clist[c]);
        float det = dltar4_pre(wv, nL, s_xka, s_xkb, s_gmk, s_rho, s_irho,
                               s_rho2, s_irho2, s_d);
        lmin = fminf(lmin, det);
        lmax = fmaxf(lmax, det);
    }
    red_min[tid] = lmin;
    red_max[tid] = lmax;
    __syncthreads();
    for (int s = 128; s > 0; s >>= 1) {
        if (tid < s) {
            red_min[tid] = fminf(red_min[tid], red_min[tid + s]);
            red_max[tid] = fmaxf(red_max[tid], red_max[tid + s]);
        }
        __syncthreads();
    }
    if (tid == 0) ws[WS_RNG + t] = red_max[0] - red_min[0];
}

// Kernel 4: eigenfunction path, compression, final scalar.
__global__ void final_kernel(const float* __restrict__ vlist,
                             float* __restrict__ ws, float* __restrict__ out,
                             int nL, int nT)
{
    __shared__ float s_ia[64], s_ib[64], s_2b2[64];
    __shared__ float s_rho[64], s_irho[64], s_rho2[64], s_irho2[64], s_d[64];
    __shared__ float red[512];
    int tid = threadIdx.x;
    for (int i = tid; i < nL; i += blockDim.x) {
        s_ia[i]    = ws[WS_INV_A + i];
        s_ib[i]    = ws[WS_INV_B + i];
        s_2b2[i]   = ws[WS_TWOB2 + i];
        s_rho[i]   = ws[WS_RHO + i];
        s_irho[i]  = ws[WS_IRHO + i];
        s_rho2[i]  = ws[WS_RHO2 + i];
        s_irho2[i] = ws[WS_IRHO2 + i];
        s_d[i]     = ws[WS_D + i];
    }
    __syncthreads();
    float val = 0.0f;
    if (tid < nT) {
        float omega = ws[WS_OMEGA + tid];
        float wv = omega * frcp(vlist[tid]);
        float det = dltar4_gen(wv, omega, nL, s_ia, s_ib, s_2b2, s_rho, s_irho,
                               s_rho2, s_irho2, s_d);
        float e = det * frcp(ws[WS_RNG + tid]);
        // 0.1^|e| - 1 == exp(ln(0.1)*|e|) - 1
        float comp = __expf(-2.302585092994046f * fabsf(e)) - 1.0f;
        val = fabsf(comp);
    }
    red[tid] = val;
    __syncthreads();
    for (int s = 256; s > 0; s >>= 1) {
        if (tid < s) red[tid] += red[tid + s];
        __syncthreads();
    }
    if (tid == 0) out[0] = red[0] / (float)nT + ws[WS_MNORM];
}

extern "C" void kernel_launch(void* const* d_in, const int* in_sizes, int n_in,
                              void* d_out, int out_size, void* d_ws, size_t ws_size,
                              hipStream_t stream)
{
    const float* vlist = (const float*)d_in[0];
    const float* tlist = (const float*)d_in[1];
    const float* b     = (const float*)d_in[2];
    const float* d     = (const float*)d_in[3];
    const float* Clist = (const float*)d_in[4];
    const float* L     = (const float*)d_in[5];
    int nT = in_sizes[1];   // 512
    int nL = in_sizes[2];   // 50
    int nC = in_sizes[4];   // 3000
    float* ws  = (float*)d_ws;
    float* out = (float*)d_out;

    prep_kernel<<<1, 512, 0, stream>>>(tlist, b, d, ws, nL, nT);
    wmma_matvec_kernel<<<1, 32, 0, stream>>>(L, b, ws, nL);
    det_kernel<<<nT, 256, 0, stream>>>(Clist, ws, nL, nC);
    final_kernel<<<1, 512, 0, stream>>>(vlist, ws, out, nL, nT);
}